// Generalized_loss_eff_old_68049461838564
// MI455X (gfx1250) — compile-verified
//
#include <hip/hip_runtime.h>
#include <hip/hip_bf16.h>
#include <stdint.h>

// ---------------------------------------------------------------------------
// Generalized loss:
//   loss = [ sum_n sum_{i,j} |e_i - e_{i+j}| * W1[i,j]
//          + sum_n sum_{i,j} (e_i - e_{i+j})^2 * W2[i,j] ] / (L*n)
// p=2 term via Gram matrix G = E^T E (exact f32) with V_WMMA_F32_16X16X4_F32,
// fed from LDS staged with gfx1250 async-to-LDS copies (ASYNCcnt path).
// p=1 term via LDS-staged VALU kernel.
// ---------------------------------------------------------------------------

#define L   96
#define NS  8192
#define KROWS   64                    // K-chunk rows staged in LDS
#define NCHUNKS (NS / KROWS)          // 128 chunks
#define NBLK    32                    // gram blocks
#define CPB     (NCHUNKS / NBLK)      // 4 chunks per block
#define NTI     (L / 16)              // 6 tiles per dimension
#define NTILES  (NTI * NTI)           // 36 tiles of 16x16
#define NWAVES  9                     // 9 waves * 4 tiles = 36
#define TPW     4                     // tiles per wave
#define CHUNK_ELEMS (KROWS * L)       // 6144 floats, contiguous in memory

typedef __attribute__((ext_vector_type(2))) float v2f;
typedef __attribute__((ext_vector_type(8))) float v8f;

// ---------------------------------------------------------------------------
// Kernel 0: zero the workspace region (G + scalar accumulator).
// ---------------------------------------------------------------------------
__global__ void zero_ws_kernel(float* ws, int n) {
    int t = blockIdx.x * blockDim.x + threadIdx.x;
    if (t < n) ws[t] = 0.0f;
}

// ---------------------------------------------------------------------------
// Kernel 1: G = E^T E, E[n][i] = yt[n][i] - yp[n][i], via WMMA f32 16x16x4.
//
// Each block (288 threads = 9 waves) owns 4 K-chunks of 64 samples. Per chunk:
//   - async-copy yt/yp chunk (contiguous 24 KB each) into LDS
//     (global_load_async_to_lds_b128, ASYNCcnt), wait, barrier
//   - every wave computes its 4 tiles: 16 WMMAs per tile from LDS reads
// C tiles accumulate in VGPRs across all 4 chunks; one atomic merge at end.
//
// ISA VGPR layouts (CDNA5 ISA):
//   A 16x4 (MxK):  lane l -> M = l&15, VGPR v holds K = 2*(l>>4) + v
//   B 4x16 (KxN):  lane l -> N = l&15, VGPR v holds K = 2*(l>>4) + v
//   C 16x16 f32:   lane l -> N = l&15, VGPR v holds M = v + 8*(l>>4)
// Since A = E^T, both A and B loads are E[r0 + {0,1}, base + (l&15)].
// ---------------------------------------------------------------------------
__global__ void __launch_bounds__(NWAVES * 32)
gram_wmma_kernel(const float* __restrict__ yt,
                 const float* __restrict__ yp,
                 float* __restrict__ G) {
    __shared__ float yts[CHUNK_ELEMS];   // 24 KB
    __shared__ float yps[CHUNK_ELEMS];   // 24 KB

    const int tid  = threadIdx.x;
    const int wave = tid >> 5;
    const int lane = tid & 31;

    const int lcol    = lane & 15;
    const int krowoff = (lane >> 4) << 1;      // 0 for lanes 0-15, 2 for 16-31

    v8f c[TPW];
#pragma unroll
    for (int t = 0; t < TPW; ++t) c[t] = (v8f){};

    for (int cc = 0; cc < CPB; ++cc) {
        const int chunk = blockIdx.x * CPB + cc;
        const unsigned chunkByteOff = (unsigned)chunk * (CHUNK_ELEMS * 4u);

        __syncthreads();   // previous chunk's LDS reads done before overwrite

        // ---- async stage: memory -> LDS, 16 B per lane-op (ASYNCcnt) ----
        for (int f = tid * 4; f < CHUNK_ELEMS; f += blockDim.x * 4) {
            unsigned gOff   = chunkByteOff + (unsigned)f * 4u;
            unsigned ldsT   = (unsigned)(uintptr_t)(&yts[f]);
            unsigned ldsP   = (unsigned)(uintptr_t)(&yps[f]);
            asm volatile("global_load_async_to_lds_b128 %0, %1, %2"
                         :: "v"(ldsT), "v"(gOff), "s"(yt) : "memory");
            asm volatile("global_load_async_to_lds_b128 %0, %1, %2"
                         :: "v"(ldsP), "v"(gOff), "s"(yp) : "memory");
        }
        // prefetch next chunk toward L2 while this one is consumed
        if (cc + 1 < CPB) {
            const char* nxt = (const char*)yt + chunkByteOff + CHUNK_ELEMS * 4u;
            __builtin_prefetch(nxt + (tid << 6), 0, 1);
        }
        asm volatile("s_wait_asynccnt 0" ::: "memory");
        __syncthreads();   // publish all waves' LDS writes

        // ---- compute: 4 tiles per wave, 16 WMMAs per tile ----
#pragma unroll
        for (int tt = 0; tt < TPW; ++tt) {
            const int tile  = wave * TPW + tt;
            const int ibase = (tile / NTI) * 16;
            const int jbase = (tile % NTI) * 16;
            const int colA  = ibase + lcol;
            const int colB  = jbase + lcol;
            v8f acc = c[tt];
            for (int kk = 0; kk < KROWS; kk += 4) {
                const int r0 = (kk + krowoff) * L;
                v2f a, b;
                a.x = yts[r0 + colA]     - yps[r0 + colA];
                a.y = yts[r0 + L + colA] - yps[r0 + L + colA];
                b.x = yts[r0 + colB]     - yps[r0 + colB];
                b.y = yts[r0 + L + colB] - yps[r0 + L + colB];
                acc = __builtin_amdgcn_wmma_f32_16x16x4_f32(
                        /*neg_a=*/false, a, /*neg_b=*/false, b,
                        /*c_mod=*/(short)0, acc,
                        /*reuse_a=*/false, /*reuse_b=*/false);
            }
            c[tt] = acc;
        }
    }

    // ---- merge: one atomic pass per block ----
    const int m0 = (lane >> 4) ? 8 : 0;
#pragma unroll
    for (int tt = 0; tt < TPW; ++tt) {
        const int tile  = wave * TPW + tt;
        const int ibase = (tile / NTI) * 16;
        const int jbase = (tile % NTI) * 16;
#pragma unroll
        for (int v = 0; v < 8; ++v) {
            const int gi = (ibase + m0 + v) * L + (jbase + lcol);
            atomicAdd(&G[gi], c[tt][v]);      // global_atomic_add_f32
        }
    }
}

// ---------------------------------------------------------------------------
// Kernel 2: p=1 term. Each wave handles one sample at a time; e-vector and W1
// staged in LDS; wave-shuffle reduction; one atomic per wave.
// grid = 256 blocks x 256 threads (8 waves); 4 samples per wave -> 8192.
// ---------------------------------------------------------------------------
__global__ void loss_p1_kernel(const float* __restrict__ yt,
                               const float* __restrict__ yp,
                               const float* __restrict__ w1,
                               float* __restrict__ acc1) {
    __shared__ float w1s[L * L];   // 36 KB
    __shared__ float es[8][L];     // 3 KB, one e-vector per wave

    for (int t = threadIdx.x; t < L * L; t += blockDim.x) w1s[t] = w1[t];

    const int wave = threadIdx.x >> 5;
    const int lane = threadIdx.x & 31;

    float a = 0.0f;
    for (int g = 0; g < 4; ++g) {
        const int s = (blockIdx.x * 32) + (g * 8) + wave;   // sample index
        __syncthreads();                                    // also covers w1s
        for (int t = lane; t < L; t += 32) {
            const int idx = s * L + t;
            es[wave][t] = yt[idx] - yp[idx];
        }
        __syncthreads();
        for (int idx = lane; idx < L * L; idx += 32) {
            const int i = idx / L;
            const int j = idx - i * L;
            const int k = i + j;
            float d;
            if (j == 0)       d = fabsf(es[wave][i]);
            else if (k < L)   d = fabsf(es[wave][i] - es[wave][k]);
            else              d = 0.0f;
            a += d * w1s[idx];
        }
    }

    // wave32 butterfly reduction
    for (int m = 16; m >= 1; m >>= 1) a += __shfl_xor(a, m, 32);
    if (lane == 0) atomicAdd(acc1, a);
}

// ---------------------------------------------------------------------------
// Kernel 3: finalize. p=2 weighted sum from G, add p=1 accumulator, scale.
//   j==0:           sum_n e_i^2              = G[i,i]
//   j>=1, i+j < L:  sum_n (e_i - e_{i+j})^2  = G[i,i] - 2 G[i,k] + G[k,k]
// ---------------------------------------------------------------------------
__global__ void finalize_kernel(const float* __restrict__ G,
                                const float* __restrict__ w2,
                                const float* __restrict__ acc1,
                                float* __restrict__ out) {
    __shared__ float red[8];
    float a = 0.0f;
    for (int idx = threadIdx.x; idx < L * L; idx += blockDim.x) {
        const int i = idx / L;
        const int j = idx - i * L;
        const int k = i + j;
        float t;
        if (j == 0)     t = G[i * L + i];
        else if (k < L) t = G[i * L + i] - 2.0f * G[i * L + k] + G[k * L + k];
        else            t = 0.0f;
        a += t * w2[idx];
    }
    for (int m = 16; m >= 1; m >>= 1) a += __shfl_xor(a, m, 32);
    const int wave = threadIdx.x >> 5;
    const int lane = threadIdx.x & 31;
    if (lane == 0) red[wave] = a;
    __syncthreads();
    if (threadIdx.x == 0) {
        float s = acc1[0];
        for (int w = 0; w < 8; ++w) s += red[w];
        out[0] = s / (float)((long)L * (long)NS);
    }
}

// ---------------------------------------------------------------------------
// Launch: zero ws -> WMMA Gram -> p1 -> finalize (all serialized on stream).
// Workspace layout: ws[0] = p1 accumulator, ws[16..16+L*L) = G (f32).
// ---------------------------------------------------------------------------
extern "C" void kernel_launch(void* const* d_in, const int* in_sizes, int n_in,
                              void* d_out, int out_size, void* d_ws, size_t ws_size,
                              hipStream_t stream) {
    const float* yt = (const float*)d_in[0];   // y_true  (8192, 96) f32
    const float* yp = (const float*)d_in[1];   // y_pred  (8192, 96) f32
    const float* w  = (const float*)d_in[2];   // weights (2, 96, 96) f32
    float* out  = (float*)d_out;               // scalar f32
    float* ws   = (float*)d_ws;
    float* acc1 = ws;                          // p1 accumulator
    float* G    = ws + 16;                     // 96*96 Gram matrix

    const int nz = 16 + L * L;
    zero_ws_kernel<<<(nz + 255) / 256, 256, 0, stream>>>(ws, nz);

    // 32 blocks x 9 waves; each block owns 4 K-chunks of 64 samples.
    gram_wmma_kernel<<<NBLK, NWAVES * 32, 0, stream>>>(yt, yp, G);

    loss_p1_kernel<<<NS / 32, 256, 0, stream>>>(yt, yp, /*W1=*/w, acc1);

    finalize_kernel<<<1, 256, 0, stream>>>(G, /*W2=*/w + L * L, acc1, out);
}